// FlashFFTConvND_13091060318307
// MI455X (gfx1250) — compile-verified
//
#include <hip/hip_runtime.h>
#include <hip/hip_bf16.h>
#include <math.h>

// ---------------------------------------------------------------------------
// FlashFFTConv 2D (256x256 FFT) for MI455X / gfx1250.
//   y = postgate * crop128( irfft2( rfft2(pregate*u) * rfft2(k) ) )
//
// Strategy:
//   Kernel 1 (kf_precompute): per-channel rfft2 of the 64x64 kernel padded to
//     256x256, stored as complex64 [C][132(col)][256(h)] in d_ws (~69 MB,
//     resident in the 192MB L2). Column-major-in-w layout so the main
//     kernel's spectral multiply reads are fully coalesced.
//   Kernel 2 (conv_main): one workgroup per (b,c) image. Entire pipeline
//     (gate, row rFFT, column FFT, spectral multiply, inverse column FFT,
//     inverse row rFFT w/ Hermitian gather, gate, store) lives in LDS.
//     HBM traffic ~= inputs + output only (~260 MB total -> ~11us bound).
//
//   All DFT butterflies are 16x16 matmuls on V_WMMA_F32_16X16X4_F32:
//   256-pt FFT = (16-pt DFT) . twiddle . (16-pt DFT), each stage = one
//   complex 16x16x16 matmul = 16 WMMA ops (4 real products x 4 K-chunks).
// ---------------------------------------------------------------------------

typedef float v2f __attribute__((ext_vector_type(2)));
typedef float v8f __attribute__((ext_vector_type(8)));

#define PI_F 3.14159265358979323846f
#define RSTRIDE 133   // float2 stride for LDS image rows (spreads LDS banks)
#define KFCOLS 132    // padded number of stored frequency columns in k_f

__device__ __forceinline__ v8f wmma_f32(v2f a, v2f b, v8f c) {
  // D = A(16x4) * B(4x16) + C ; f32 WMMA: no A/B negate (ISA NEG = CNeg only)
  return __builtin_amdgcn_wmma_f32_16x16x4_f32(false, a, false, b, (short)0, c,
                                               false, false);
}

// A-matrix 16x4 fragment (ISA 7.12.2): lanes 0-15 hold K=kc,kc+1 for M=lane;
// lanes 16-31 hold K=kc+2,kc+3 for M=lane-16.
template <typename F>
__device__ __forceinline__ v2f load_a_frag(F f, int kc, int lane) {
  const int r  = lane & 15;
  const int hi = (lane >> 4) & 1;
  v2f a;
  a.x = f(r, kc + 2 * hi);
  a.y = f(r, kc + 2 * hi + 1);
  return a;
}

// B-matrix 4x16 fragment: rows striped across lanes within a VGPR:
// v0: lanes0-15 -> K=kc, lanes16-31 -> K=kc+2 ; v1: K=kc+1 / K=kc+3.
template <typename F>
__device__ __forceinline__ v2f load_b_frag(F f, int kc, int lane) {
  const int c  = lane & 15;
  const int hi = (lane >> 4) & 1;
  v2f b;
  b.x = f(kc + 2 * hi, c);
  b.y = f(kc + 2 * hi + 1, c);
  return b;
}

// Complex 16x16x16 matmul: (Dre + i*Dim) = (Ar + i*Ai) * (Br + i*Bi).
// 4 real accumulators, 16 v_wmma_f32_16x16x4_f32 total.
template <typename FAr, typename FAi, typename FBr, typename FBi>
__device__ __forceinline__ void cmatmul16(FAr ar_f, FAi ai_f, FBr br_f,
                                          FBi bi_f, int lane, v8f& dre,
                                          v8f& dim) {
  v8f rr = {}, ii = {}, ri = {}, ir = {};
#pragma unroll
  for (int kc = 0; kc < 16; kc += 4) {
    v2f ar = load_a_frag(ar_f, kc, lane);
    v2f ai = load_a_frag(ai_f, kc, lane);
    v2f br = load_b_frag(br_f, kc, lane);
    v2f bi = load_b_frag(bi_f, kc, lane);
    rr = wmma_f32(ar, br, rr);
    ii = wmma_f32(ai, bi, ii);
    ri = wmma_f32(ar, bi, ri);
    ir = wmma_f32(ai, br, ir);
  }
  dre = rr - ii;
  dim = ri + ir;
}

// 256-point complex FFT of one sequence, radix 16x16, executed by ONE wave.
//   n = n1 + 16*n2, k = k1 + 16*k2
//   stage1: B[n1][k1]   = sum_n2 src(n1+16*n2) * F16[n2][k1]      (X * F)
//   twiddle: B'[n1][k1] = B[n1][k1] * W256^(n1*k1)
//   stage2: C[k2][k1]   = sum_n1 F16[k2][n1] * B'[n1][k1]         (F * B')
//   dst(k1 + 16*k2) = C[k2][k1]
// dir = +1 forward (tables hold e^{-i t}), dir = -1 inverse (conjugate).
template <typename SRC, typename DST>
__device__ __forceinline__ void fft256(SRC src, DST dst, const float2* Ftab,
                                       const float2* Wtab, float2* spill,
                                       float dir, int lane) {
  v8f bre, bim;
  cmatmul16(
      [&](int r, int c) { return src(r + 16 * c).x; },
      [&](int r, int c) { return src(r + 16 * c).y; },
      [&](int r, int c) { return Ftab[r * 16 + c].x; },
      [&](int r, int c) { return dir * Ftab[r * 16 + c].y; }, lane, bre, bim);

  // Twiddle in D-register layout (reg j -> row j + 8*hi, col = lane&15),
  // then spill transposed operand for stage 2 through per-wave LDS.
  const int col   = lane & 15;
  const int rbase = (lane >> 4) ? 8 : 0;
#pragma unroll
  for (int j = 0; j < 8; ++j) {
    const int row = j + rbase;
    const float2 w = Wtab[(row * col) & 255];
    const float wr = w.x, wi = dir * w.y;
    const float re = bre[j] * wr - bim[j] * wi;
    const float im = bre[j] * wi + bim[j] * wr;
    spill[row * 16 + col] = make_float2(re, im);
  }
  // Same-wave LDS RAW across lanes: LDS is in-order per wave; make explicit.
  asm volatile("s_wait_dscnt 0" ::: "memory");

  v8f cre, cim;
  cmatmul16(
      [&](int r, int c) { return Ftab[r * 16 + c].x; },
      [&](int r, int c) { return dir * Ftab[r * 16 + c].y; },
      [&](int r, int c) { return spill[r * 16 + c].x; },
      [&](int r, int c) { return spill[r * 16 + c].y; }, lane, cre, cim);

#pragma unroll
  for (int j = 0; j < 8; ++j) {
    const int k2 = j + rbase;
    dst(col + 16 * k2, make_float2(cre[j], cim[j]));
  }
}

__device__ __forceinline__ void init_tables(float2* Ftab, float2* Wtab,
                                            int tid) {
  for (int i = tid; i < 256; i += 256) {
    const int r = i >> 4, c = i & 15;
    const float tf = -2.0f * PI_F * (float)((r * c) & 15) / 16.0f;
    Ftab[i] = make_float2(cosf(tf), sinf(tf));
    const float tw = -2.0f * PI_F * (float)i / 256.0f;
    Wtab[i] = make_float2(cosf(tw), sinf(tw));
  }
}

// ---------------------------------------------------------------------------
// Kernel 1: k_f[c] = rfft2(pad256(k[c])).
// Layout: kf[(c*KFCOLS + col)*256 + h]  (contiguous in h for coalescing).
// ---------------------------------------------------------------------------
__global__ void __launch_bounds__(256) kf_precompute(
    const float* __restrict__ kin, float2* __restrict__ kf) {
  extern __shared__ char smem[];
  float2* Rk      = (float2*)smem;                       // [64][RSTRIDE]
  float2* Ftab    = (float2*)(smem + 64 * RSTRIDE * 8);  // [256]
  float2* Wtab    = Ftab + 256;                          // [256]
  float2* scratch = Wtab + 256;                          // [8][512]

  const int tid = threadIdx.x, lane = tid & 31, wave = tid >> 5;
  const int c = blockIdx.x;

  init_tables(Ftab, Wtab, tid);
  const float* kc = kin + (size_t)c * (64 * 64);
  for (int i = tid; i < 64 * 64; i += 256) {
    const int r = i >> 6, w = i & 63;
    Rk[r * RSTRIDE + w] = make_float2(kc[i], 0.0f);
  }
  __syncthreads();

  float2* spill = scratch + wave * 512;

  // Row rFFTs: 64 real rows (zero-padded to 256), keep bins 0..128, in place.
  for (int row = wave; row < 64; row += 8) {
    float2* Rrow = Rk + row * RSTRIDE;
    fft256(
        [&](int n) {
          return (n < 64) ? make_float2(Rrow[n].x, 0.0f) : make_float2(0.f, 0.f);
        },
        [&](int k, float2 v) { if (k <= 128) Rrow[k] = v; },
        Ftab, Wtab, spill, +1.0f, lane);
  }
  __syncthreads();

  // Column FFTs for 129 frequency columns; write full 256 bins to global,
  // one contiguous 2KB block per column.
  for (int col = wave; col < 129; col += 8) {
    float2* kfcol = kf + ((size_t)c * KFCOLS + col) * 256;
    fft256(
        [&](int n) {
          return (n < 64) ? Rk[n * RSTRIDE + col] : make_float2(0.f, 0.f);
        },
        [&](int k, float2 v) { kfcol[k] = v; },
        Ftab, Wtab, spill, +1.0f, lane);
  }
}

// ---------------------------------------------------------------------------
// Kernel 2: fully fused per-image conv. One block per (b,c).
// ---------------------------------------------------------------------------
__global__ void __launch_bounds__(256) conv_main(
    const float* __restrict__ u, const float* __restrict__ pregate,
    const float* __restrict__ postgate, const float2* __restrict__ kf,
    float* __restrict__ y) {
  extern __shared__ char smem[];
  float2* R       = (float2*)smem;                        // [128][RSTRIDE]
  float2* Ftab    = (float2*)(smem + 128 * RSTRIDE * 8);  // [256]
  float2* Wtab    = Ftab + 256;                           // [256]
  float2* scratch = Wtab + 256;                           // [8][512]

  const int tid = threadIdx.x, lane = tid & 31, wave = tid >> 5;
  const int img = blockIdx.x;  // b * 256 + c  (b-major == memory layout)
  const int c = img & 255;
  const size_t base = (size_t)img * (128 * 128);

  init_tables(Ftab, Wtab, tid);

  // Stage gated input into real part of R (cols 0..127).
  for (int i = tid; i < 128 * 128; i += 256) {
    const int r = i >> 7, w = i & 127;
    R[r * RSTRIDE + w] = make_float2(u[base + i] * pregate[base + i], 0.0f);
  }
  __syncthreads();

  float2* spill = scratch + wave * 512;
  float2* spec  = spill + 256;

  // -------- Forward row rFFTs (128 rows, keep 129 bins, in place) --------
  for (int row = wave; row < 128; row += 8) {
    float2* Rrow = R + row * RSTRIDE;
    fft256(
        [&](int n) {
          return (n < 128) ? make_float2(Rrow[n].x, 0.0f)
                           : make_float2(0.f, 0.f);
        },
        [&](int k, float2 v) { if (k <= 128) Rrow[k] = v; },
        Ftab, Wtab, spill, +1.0f, lane);
  }
  __syncthreads();

  // -------- Column FFT -> spectral multiply -> inverse column FFT --------
  // Waves own disjoint column sets -> no block barrier needed inside loop.
  const float2* kfbase = kf + (size_t)c * KFCOLS * 256;
  for (int col = wave; col < 129; col += 8) {
    // Prefetch next column's 2KB k_f block (lanes spread 64B apart) so the
    // L2 fetch overlaps the two column FFTs below (global_prefetch_b8).
    if (col + 8 < 129) {
      const char* pf = (const char*)(kfbase + (size_t)(col + 8) * 256);
      __builtin_prefetch(pf + lane * 64, 0, 1);
    }

    fft256(
        [&](int n) {
          return (n < 128) ? R[n * RSTRIDE + col] : make_float2(0.f, 0.f);
        },
        [&](int k, float2 v) { spec[k] = v; },
        Ftab, Wtab, spill, +1.0f, lane);

    // Spectral multiply: coalesced 8B/lane reads from the h-contiguous k_f.
    const float2* kfcol = kfbase + (size_t)col * 256;
    for (int h = lane; h < 256; h += 32) {
      const float2 a = spec[h];
      const float2 b = kfcol[h];
      spec[h] = make_float2(a.x * b.x - a.y * b.y, a.x * b.y + a.y * b.x);
    }
    asm volatile("s_wait_dscnt 0" ::: "memory");

    fft256(
        [&](int n) { return spec[n]; },
        [&](int k, float2 v) { if (k < 128) R[k * RSTRIDE + col] = v; },
        Ftab, Wtab, spill, -1.0f, lane);
  }
  __syncthreads();

  // -------- Inverse row rFFT (Hermitian gather) + postgate + store --------
  const float inv_scale = 1.0f / 65536.0f;  // 1/(256*256) for irfft2
  for (int row = wave; row < 128; row += 8) {
    float2* Rrow = R + row * RSTRIDE;
    const size_t obase = base + (size_t)row * 128;
    fft256(
        [&](int n) {
          if (n <= 128) return Rrow[n];
          const float2 v = Rrow[256 - n];
          return make_float2(v.x, -v.y);  // Hermitian extension
        },
        [&](int k, float2 v) {
          if (k < 128) y[obase + k] = v.x * inv_scale * postgate[obase + k];
        },
        Ftab, Wtab, spill, -1.0f, lane);
  }
}

// ---------------------------------------------------------------------------
extern "C" void kernel_launch(void* const* d_in, const int* in_sizes, int n_in,
                              void* d_out, int out_size, void* d_ws,
                              size_t ws_size, hipStream_t stream) {
  const float* u        = (const float*)d_in[0];
  const float* k        = (const float*)d_in[1];
  const float* pregate  = (const float*)d_in[2];
  const float* postgate = (const float*)d_in[3];
  float* y   = (float*)d_out;
  float2* kf = (float2*)d_ws;  // needs 256*KFCOLS*256*8 = ~69 MB (L2-resident)

  const size_t lds_kf   = (size_t)64 * RSTRIDE * 8 + 512 * 8 + 8 * 512 * 8;
  const size_t lds_main = (size_t)128 * RSTRIDE * 8 + 512 * 8 + 8 * 512 * 8;

  (void)hipFuncSetAttribute((const void*)kf_precompute,
                            hipFuncAttributeMaxDynamicSharedMemorySize,
                            (int)lds_kf);
  (void)hipFuncSetAttribute((const void*)conv_main,
                            hipFuncAttributeMaxDynamicSharedMemorySize,
                            (int)lds_main);

  kf_precompute<<<256, 256, lds_kf, stream>>>(k, kf);
  conv_main<<<4 * 256, 256, lds_main, stream>>>(u, pregate, postgate, kf, y);
}